// PartInitMinimal_68710886801956
// MI455X (gfx1250) — compile-verified
//
#include <hip/hip_runtime.h>
#include <hip/hip_bf16.h>

typedef __attribute__((ext_vector_type(2))) float v2f;
typedef __attribute__((ext_vector_type(8))) float v8f;

#define N_PART_C   131072
#define N_PAIR_C   8388608
#define N_TERMS_C  20
#define OUT_CH_C   128
#define NUM_JETS_C 2048
#define EPS_C      1e-5f

// ---------------- K0: zero scratch accumulators ----------------
__global__ void zero_ws_kernel(float* __restrict__ ws, int n) {
    int i = blockIdx.x * blockDim.x + threadIdx.x;
    if (i < n) ws[i] = 0.0f;
}

// ---------------- K1: pair gather-multiply-scatter ----------------
// src[p] = part_weight[tail[p]] * pair_func[p][:]; atomic-add into part_features[head[p]][:]
// pair_func rows are 80B and start 16B-aligned -> 5x float4 loads (dominant 671MB stream).
// part_features is 10.5MB -> all 168M f32 atomics resolve in L2 (192MB).
__global__ void __launch_bounds__(256) pair_scatter_kernel(
    const float* __restrict__ part_weight,
    const int*   __restrict__ pair_head,
    const int*   __restrict__ pair_tail,
    const float* __restrict__ pair_func,
    float*       __restrict__ part_feat) {
    int p = blockIdx.x * blockDim.x + threadIdx.x;   // grid exactly covers N_PAIR
    float w = part_weight[pair_tail[p]];
    int   h = pair_head[p];
    const float4* row = reinterpret_cast<const float4*>(pair_func + (size_t)p * N_TERMS_C);
    float* dst = part_feat + (size_t)h * N_TERMS_C;
    #pragma unroll
    for (int q = 0; q < 5; ++q) {
        float4 f = row[q];
        atomicAdd(dst + 4 * q + 0, w * f.x);
        atomicAdd(dst + 4 * q + 1, w * f.y);
        atomicAdd(dst + 4 * q + 2, w * f.z);
        atomicAdd(dst + 4 * q + 3, w * f.w);
    }
}

// ---------------- K2: weighted segment-sum into jet features ----------------
__global__ void __launch_bounds__(256) jet_reduce_kernel(
    const float* __restrict__ part_feat,
    const float* __restrict__ part_weight,
    const int*   __restrict__ part_indicator,
    float*       __restrict__ jet_feat) {
    int i = blockIdx.x * blockDim.x + threadIdx.x;   // grid exactly covers N_PART
    float w = part_weight[i];
    int   j = part_indicator[i];
    const float* src = part_feat + (size_t)i * N_TERMS_C;
    float*       dst = jet_feat  + (size_t)j * N_TERMS_C;
    #pragma unroll
    for (int t = 0; t < N_TERMS_C; ++t)
        atomicAdd(dst + t, src[t] * w);
}

// ---------------- K3: mean/var (ddof=1) + fold BN into FC weights/bias ----------------
__global__ void __launch_bounds__(256) stats_fold_kernel(
    const float* __restrict__ jet_feat,
    const float* __restrict__ fc_w,
    const float* __restrict__ fc_b,
    float*       __restrict__ wprime,   // [OUT_CH][N_TERMS]
    float*       __restrict__ bprime) { // [OUT_CH]
    __shared__ float smean[N_TERMS_C];
    __shared__ float sistd[N_TERMS_C];
    int tid = threadIdx.x;
    if (tid < N_TERMS_C) {
        float s = 0.0f, ss = 0.0f;
        for (int j = 0; j < NUM_JETS_C; ++j) {
            float v = jet_feat[(size_t)j * N_TERMS_C + tid];
            s += v; ss += v * v;
        }
        float mean = s / (float)NUM_JETS_C;
        float var  = (ss - (float)NUM_JETS_C * mean * mean) / (float)(NUM_JETS_C - 1);
        smean[tid] = mean;
        sistd[tid] = rsqrtf(var + EPS_C);
    }
    __syncthreads();
    for (int i = tid; i < OUT_CH_C * N_TERMS_C; i += blockDim.x) {
        int t = i % N_TERMS_C;
        wprime[i] = fc_w[i] * sistd[t];
    }
    if (tid < OUT_CH_C) {
        float b = fc_b[tid];
        #pragma unroll
        for (int t = 0; t < N_TERMS_C; ++t)
            b -= smean[t] * sistd[t] * fc_w[tid * N_TERMS_C + t];
        bprime[tid] = b;
    }
}

// ---------------- K4: out = part_feat @ wprime^T + bprime via V_WMMA_F32_16X16X4_F32 ----
// Block = 256 threads = 8 waves. All waves share rows [16*blockIdx, +16); wave w owns
// channels [16*w, +16). K=20 -> 5 WMMA steps of K=4.
// A 16x4 layout: lane<16 holds K={kb,kb+1}, lane>=16 holds K={kb+2,kb+3}, M=lane&15.
// B 4x16 mirrored (N=lane&15). C/D: reg r -> row r+8*(lane>>4), col lane&15.
__global__ void __launch_bounds__(256) gemm_wmma_kernel(
    const float* __restrict__ part_feat,
    const float* __restrict__ wprime,
    const float* __restrict__ bprime,
    float*       __restrict__ out) {
    int lane = threadIdx.x & 31;
    int wave = threadIdx.x >> 5;
    int lm   = lane & 15;
    int hi   = lane >> 4;
    int rowBase = blockIdx.x * 16;
    int chBase  = wave * 16;

    const float* arow = part_feat + (size_t)(rowBase + lm) * N_TERMS_C + 2 * hi;
    const float* brow = wprime    + (size_t)(chBase  + lm) * N_TERMS_C + 2 * hi;

    float bias = bprime[chBase + lm];   // D column (= out channel) fixed per lane
    v8f c = {bias, bias, bias, bias, bias, bias, bias, bias};

    #pragma unroll
    for (int kb = 0; kb < N_TERMS_C; kb += 4) {
        v2f a, b;
        a.x = arow[kb]; a.y = arow[kb + 1];
        b.x = brow[kb]; b.y = brow[kb + 1];
        c = __builtin_amdgcn_wmma_f32_16x16x4_f32(
                /*neg_a=*/false, a, /*neg_b=*/false, b,
                /*c_mod=*/(short)0, c, /*reuse_a=*/false, /*reuse_b=*/false);
    }

    #pragma unroll
    for (int r = 0; r < 8; ++r)
        out[(size_t)(rowBase + r + 8 * hi) * OUT_CH_C + chBase + lm] = c[r];
}

extern "C" void kernel_launch(void* const* d_in, const int* in_sizes, int n_in,
                              void* d_out, int out_size, void* d_ws, size_t ws_size,
                              hipStream_t stream) {
    (void)in_sizes; (void)n_in; (void)out_size; (void)ws_size;
    const float* part_weight    = (const float*)d_in[0];
    const int*   pair_head      = (const int*)  d_in[1];
    const int*   pair_tail      = (const int*)  d_in[2];
    const float* pair_func      = (const float*)d_in[3];
    const int*   part_indicator = (const int*)  d_in[4];
    const float* fc_w           = (const float*)d_in[5];
    const float* fc_b           = (const float*)d_in[6];
    float* out = (float*)d_out;

    // workspace layout (floats): part_feat | jet_feat | wprime | bprime
    float* ws = (float*)d_ws;
    float* part_feat = ws;
    float* jet_feat  = part_feat + (size_t)N_PART_C   * N_TERMS_C;  // 2,621,440
    float* wprime    = jet_feat  + (size_t)NUM_JETS_C * N_TERMS_C;  // +40,960
    float* bprime    = wprime    + (size_t)OUT_CH_C   * N_TERMS_C;  // +2,560

    int nzero = N_PART_C * N_TERMS_C + NUM_JETS_C * N_TERMS_C;
    zero_ws_kernel<<<(nzero + 255) / 256, 256, 0, stream>>>(ws, nzero);

    pair_scatter_kernel<<<N_PAIR_C / 256, 256, 0, stream>>>(
        part_weight, pair_head, pair_tail, pair_func, part_feat);

    jet_reduce_kernel<<<N_PART_C / 256, 256, 0, stream>>>(
        part_feat, part_weight, part_indicator, jet_feat);

    stats_fold_kernel<<<1, 256, 0, stream>>>(jet_feat, fc_w, fc_b, wprime, bprime);

    gemm_wmma_kernel<<<N_PART_C / 16, 256, 0, stream>>>(part_feat, wprime, bprime, out);
}